// Pool3d_10763188043856
// MI455X (gfx1250) — compile-verified
//
#include <hip/hip_runtime.h>

// Pooling by scatter-max: out[seg, c] = max over rows i with seg(i)==seg of in[i, c],
// empty segments -> 0.  seg(i) = vt_map[vt_replace[i]].
//
// Strategy (MI455X / gfx1250):
//  - Pure memory-bound op (~648 MB compulsory traffic, ~28us @ 23.3 TB/s). No WMMA
//    applies (no matrix math); the CDNA5 lever is the vmem/atomic/L2 system.
//  - Reduction via native no-return GLOBAL_ATOMIC_MAX_U32 on an order-preserving
//    u32 encoding of f32 (branchless, no CAS). All atomics resolve in the 192MB L2:
//    the whole 128MB output stays L2-resident (write amplification never hits HBM).
//  - 512MB input stream uses non-temporal loads (TH=NT) so it doesn't evict the
//    output working set from L2.
//  - One wave32 per input row: 32 lanes x float4 = 512B coalesced row load;
//    wave-uniform indices scalarized with readfirstlane so the second index
//    gather and the atomic base address live on the SALU/SMEM path.

typedef float    f32x4 __attribute__((ext_vector_type(4)));
typedef unsigned u32x4 __attribute__((ext_vector_type(4)));

#define CCH 128  // feature channels (per reference)

// Order-preserving map: f32 total order -> u32 total order.
// key = bits ^ ( (bits negative) ? 0xFFFFFFFF : 0x80000000 )
__device__ __forceinline__ unsigned enc_f32(float f) {
    unsigned b = __float_as_uint(f);
    return b ^ (unsigned)(((int)b >> 31) | (int)0x80000000);
}

__device__ __forceinline__ float dec_key(unsigned k) {
    // Inverse of enc_f32; key==0 decodes to 0xFFFFFFFF (-NaN) -> mapped to 0.
    unsigned b = (k & 0x80000000u) ? (k ^ 0x80000000u) : ~k;
    // non-finite (empty-segment sentinel, or inf/nan max) -> 0, matching
    // jnp.where(isfinite(out), out, 0)
    return ((b & 0x7F800000u) == 0x7F800000u) ? 0.0f : __uint_as_float(b);
}

// ---------------------------------------------------------------------------
// Pass 1: fill output key buffer with 0 (the minimum key). 128-bit stores.
__global__ void __launch_bounds__(256) seg_init(u32x4* __restrict__ out4, int n4) {
    int i = blockIdx.x * blockDim.x + threadIdx.x;
    if (i < n4) {
        u32x4 z = {0u, 0u, 0u, 0u};
        out4[i] = z;
    }
}

// ---------------------------------------------------------------------------
// Pass 2: scatter-max. One wave per input row; lane L owns channels 4L..4L+3.
__global__ void __launch_bounds__(256) seg_scatter(const f32x4* __restrict__ in4,
                                                   const int*   __restrict__ vt_replace,
                                                   const int*   __restrict__ vt_map,
                                                   unsigned*    __restrict__ out,
                                                   int n_rows) {
    int t     = blockIdx.x * blockDim.x + threadIdx.x;
    int row   = t >> 5;        // wave-uniform (blockDim multiple of 32)
    int lane4 = t & 31;
    if (row >= n_rows) return; // whole wave exits together (row is uniform)

    // First gather: wave-uniform address (L0 merges the 32 identical requests).
    int rep = vt_replace[row];
    // Scalarize so the second gather has a provably uniform (SGPR) address ->
    // eligible for s_load_b32 (SMEM path), and downstream math stays on SALU.
    rep = __builtin_amdgcn_readfirstlane(rep);
    int seg = vt_map[rep];
    seg = __builtin_amdgcn_readfirstlane(seg);

    // Streaming read: non-temporal so the 512MB input doesn't evict the
    // L2-resident output. 32 lanes x 16B = full 512B row, one b128 per lane.
    f32x4 v = __builtin_nontemporal_load(&in4[(size_t)row * (CCH / 4) + lane4]);

    unsigned* p = out + (size_t)seg * CCH + (lane4 << 2);
    // No-return GLOBAL_ATOMIC_MAX_U32 (result discarded) -> STOREcnt path, L2 RMW.
    atomicMax(p + 0, enc_f32(v.x));
    atomicMax(p + 1, enc_f32(v.y));
    atomicMax(p + 2, enc_f32(v.z));
    atomicMax(p + 3, enc_f32(v.w));
}

// ---------------------------------------------------------------------------
// Pass 3: decode keys in place: key -> float, non-finite -> 0. 128-bit ld/st.
__global__ void __launch_bounds__(256) seg_finalize(u32x4* __restrict__ io4, int n4) {
    int i = blockIdx.x * blockDim.x + threadIdx.x;
    if (i >= n4) return;
    u32x4 k = io4[i];
    f32x4 r;
    r.x = dec_key(k.x);
    r.y = dec_key(k.y);
    r.z = dec_key(k.z);
    r.w = dec_key(k.w);
    *((f32x4*)&io4[i]) = r;
}

// ---------------------------------------------------------------------------
extern "C" void kernel_launch(void* const* d_in, const int* in_sizes, int n_in,
                              void* d_out, int out_size, void* d_ws, size_t ws_size,
                              hipStream_t stream) {
    const float* inputs     = (const float*)d_in[0];  // [N_IN, 128] f32
    const int*   vt_replace = (const int*)d_in[1];    // [N_IN] i32
    const int*   vt_map     = (const int*)d_in[2];    // [N_IN] i32
    // d_in[3] (vt_out scalar) not needed on host: out_size == N_OUT * 128.

    const int n_rows = in_sizes[1];       // N_IN
    const int n4     = out_size / 4;      // output elements / 4 (128-bit chunks)

    // Pass 1: init keys to 0 (minimum of the encoding).
    seg_init<<<(n4 + 255) / 256, 256, 0, stream>>>((u32x4*)d_out, n4);

    // Pass 2: scatter-max, 32 threads (one wave) per input row.
    long long tasks  = (long long)n_rows * 32;
    unsigned  blocks = (unsigned)((tasks + 255) / 256);
    seg_scatter<<<blocks, 256, 0, stream>>>((const f32x4*)inputs, vt_replace, vt_map,
                                            (unsigned*)d_out, n_rows);

    // Pass 3: decode in place.
    seg_finalize<<<(n4 + 255) / 256, 256, 0, stream>>>((u32x4*)d_out, n4);
}